// RegularTensorProductLayer_34308198761119
// MI455X (gfx1250) — compile-verified
//
#include <hip/hip_runtime.h>
#include <hip/hip_bf16.h>
#include <math.h>

typedef float v2f __attribute__((ext_vector_type(2)));
typedef float v8f __attribute__((ext_vector_type(8)));

#define NPATH 23
#define ZT 16          // samples per workgroup (= WMMA N)
#define THREADS 256    // 8 waves (wave32)
#define S_X 1028       // XL row stride (floats): 16B-aligned rows, bank stride 4
#define WLSTRIDE 4160  // one WL buffer: 64*65 floats

// Path table, lexicographic over (l1,l2,l3), e3nn FullyConnectedTensorProduct order.
__constant__ int PL1[NPATH] = {0,0,0,0, 1,1,1,1,1,1, 2,2,2,2,2,2,2, 3,3,3,3,3,3};
__constant__ int PL2[NPATH] = {0,1,2,3, 0,1,1,2,2,3, 0,1,1,2,2,3,3, 0,1,2,2,3,3};
__constant__ int PL3[NPATH] = {0,1,2,3, 1,0,2,1,3,2, 2,1,3,0,2,1,3, 3,2,1,3,0,2};
__constant__ int W3J_OFF[NPATH] = {0,1,10,35, 84,93,102,147,192,297,
                                   402,427,472,577,602,727,832,
                                   1077,1126,1231,1336,1581,1630};
__constant__ int NTO[4] = {4,6,7,6};

// ---------------------------------------------------------------------------
// CDNA5 async global->LDS (ASYNCcnt). lds_off = byte offset inside the
// workgroup LDS allocation (low 32 bits of a generic pointer to __shared__).
// ---------------------------------------------------------------------------
__device__ __forceinline__ void async_load_b128(unsigned lds_off, const void* g) {
  asm volatile("global_load_async_to_lds_b128 %0, %1, off"
               :: "v"(lds_off), "v"((unsigned long long)(uintptr_t)g)
               : "memory");
}
__device__ __forceinline__ void async_load_b32(unsigned lds_off, const void* g) {
  asm volatile("global_load_async_to_lds_b32 %0, %1, off"
               :: "v"(lds_off), "v"((unsigned long long)(uintptr_t)g)
               : "memory");
}
__device__ __forceinline__ void wait_async0()  { asm volatile("s_wait_asynccnt 0"  ::: "memory"); }
__device__ __forceinline__ void wait_async16() { asm volatile("s_wait_asynccnt 16" ::: "memory"); }

// Scatter one 16KB weight slab (W_p[u]) into a WL buffer, transposed to
// WL[w*65 + v]; the async engine performs the transpose via per-lane dsaddr.
__device__ __forceinline__ void issue_wl(const float* g4096, const float* wlbuf, int tid) {
  const unsigned base = (unsigned)(uintptr_t)wlbuf;
#pragma unroll
  for (int it = 0; it < 16; ++it) {
    int e = it * THREADS + tid;
    int v = e >> 6, ww = e & 63;
    async_load_b32(base + (unsigned)((ww * 65 + v) * 4), g4096 + e);
  }
}

// ---------------------------------------------------------------------------
// Init kernel: real-basis Wigner 3j tensors (e3nn convention) in fp64.
// ---------------------------------------------------------------------------
__device__ double dfact(int n) {
  double r = 1.0;
  for (int i = 2; i <= n; ++i) r *= (double)i;
  return r;
}

__device__ double su2_cg(int j1, int m1, int j2, int m2, int j3, int m3) {
  if (m3 != m1 + m2) return 0.0;
  int vmin = -j1 + j2 + m3; if (-j1 + m1 > vmin) vmin = -j1 + m1; if (0 > vmin) vmin = 0;
  int vmax = j2 + j3 + m1; if (j3 - j1 + j2 < vmax) vmax = j3 - j1 + j2; if (j3 + m3 < vmax) vmax = j3 + m3;
  double c = sqrt((2.0*j3 + 1.0) * dfact(j3 + j1 - j2) * dfact(j3 - j1 + j2) * dfact(j1 + j2 - j3)
                  * dfact(j3 + m3) * dfact(j3 - m3)
                  / (dfact(j1 + j2 + j3 + 1) * dfact(j1 - m1) * dfact(j1 + m1)
                     * dfact(j2 - m2) * dfact(j2 + m2)));
  double s = 0.0;
  for (int v = vmin; v <= vmax; ++v) {
    double sign = ((v + j2 + m2) & 1) ? -1.0 : 1.0;
    s += sign / dfact(v) * dfact(j2 + j3 + m1 - v) * dfact(j1 - m1 + v)
         / dfact(j3 - j1 + j2 - v) / dfact(j3 + m3 - v) / dfact(v + j1 - j2 - m3);
  }
  return c * s;
}

__device__ void build_q(int l, double* qre, double* qim) {
  int d = 2 * l + 1;
  for (int i = 0; i < d * d; ++i) { qre[i] = 0.0; qim[i] = 0.0; }
  double is2 = 1.0 / sqrt(2.0);
  for (int m = -l; m < 0; ++m) {
    qre[(l + m) * d + (l - m)] = is2;
    qim[(l + m) * d + (l + m)] = -is2;
  }
  qre[l * d + l] = 1.0;
  for (int m = 1; m <= l; ++m) {
    double sg = (m & 1) ? -1.0 : 1.0;
    qre[(l + m) * d + (l + m)] = sg * is2;
    qim[(l + m) * d + (l - m)] = sg * is2;
  }
  int ph = l & 3;
  for (int i = 0; i < d * d; ++i) {
    double re = qre[i], im = qim[i];
    if (ph == 1)      { qre[i] = im;  qim[i] = -re; }
    else if (ph == 2) { qre[i] = -re; qim[i] = -im; }
    else if (ph == 3) { qre[i] = -im; qim[i] = re; }
  }
}

__global__ void w3j_init_kernel(float* __restrict__ w3j_out) {
  int p = threadIdx.x;
  if (p >= NPATH) return;
  int l1 = PL1[p], l2 = PL2[p], l3 = PL3[p];
  int d1 = 2 * l1 + 1, d2 = 2 * l2 + 1, d3 = 2 * l3 + 1;

  double csu2[7][7][7];
  for (int i = 0; i < 7; ++i)
    for (int k = 0; k < 7; ++k)
      for (int n = 0; n < 7; ++n) csu2[i][k][n] = 0.0;
  for (int m1 = -l1; m1 <= l1; ++m1)
    for (int m2 = -l2; m2 <= l2; ++m2) {
      int m3 = m1 + m2;
      if (m3 >= -l3 && m3 <= l3)
        csu2[l1 + m1][l2 + m2][l3 + m3] = su2_cg(l1, m1, l2, m2, l3, m3);
    }

  double q1re[49], q1im[49], q2re[49], q2im[49], q3re[49], q3im[49];
  build_q(l1, q1re, q1im);
  build_q(l2, q2re, q2im);
  build_q(l3, q3re, q3im);

  double cc[343];
  double nrm2 = 0.0;
  for (int j = 0; j < d1; ++j)
    for (int l = 0; l < d2; ++l)
      for (int m = 0; m < d3; ++m) {
        double ar_sum = 0.0;
        for (int i = 0; i < d1; ++i)
          for (int k = 0; k < d2; ++k)
            for (int n = 0; n < d3; ++n) {
              double s = csu2[i][k][n];
              if (s == 0.0) continue;
              double ar = q1re[i * d1 + j], ai = q1im[i * d1 + j];
              double br = q2re[k * d2 + l], bi = q2im[k * d2 + l];
              double zr = ar * br - ai * bi;
              double zi = ar * bi + ai * br;
              ar_sum += (zr * q3re[n * d3 + m] + zi * q3im[n * d3 + m]) * s;
            }
        cc[(j * d2 + l) * d3 + m] = ar_sum;
        nrm2 += ar_sum * ar_sum;
      }
  double inv = 1.0 / sqrt(nrm2);
  int sz = d1 * d2 * d3;
  for (int i = 0; i < sz; ++i)
    w3j_out[W3J_OFF[p] + i] = (float)(cc[i] * inv);
}

// ---------------------------------------------------------------------------
// Per-path body, fully specialized on (L1,L2,L3).
// ---------------------------------------------------------------------------
template<int L1, int L2, int L3>
__device__ void run_path(int p, int zbase, int tid, int lane, int mth, int kq,
                         const float* __restrict__ wts,
                         const float* __restrict__ w3j,
                         float* __restrict__ outp,
                         float* XL, float* WLB, float* T2L, float* T1L, float* CW) {
  constexpr int D1 = 2*L1 + 1, D2 = 2*L2 + 1, KK = 2*L3 + 1;
  constexpr int OFF1 = 64*L1*L1, OFF2 = 64*L2*L2, OFF3 = 64*L3*L3;
  constexpr int JK = D2 * KK;
  constexpr int NTO_C[4] = {4, 6, 7, 6};
  const float coeff = sqrtf((float)KK / ((float)NTO_C[L3] * 4096.0f));

  // w3j of this path into LDS
  for (int i = tid; i < D1*D2*KK; i += THREADS) CW[i] = w3j[W3J_OFF[p] + i];

  v8f acc[2][2];
#pragma unroll
  for (int t = 0; t < 2; ++t)
#pragma unroll
    for (int q = 0; q < 2; ++q)
      acc[t][q] = (v8f){0.f,0.f,0.f,0.f,0.f,0.f,0.f,0.f};
  __syncthreads();   // CW ready

  for (int u = 0; u < 64; ++u) {
    // ---- T1[z,j,k] = sum_i x1[z,u,i] * C[i,j,k] ----
    for (int e = tid; e < ZT*JK; e += THREADS) {
      int z = e / JK;
      int r = e - z*JK;
      int j = r / KK;
      int k = r - j*KK;
      const float* xz = &XL[z*S_X + OFF1 + u*D1];
      float s = 0.0f;
#pragma unroll
      for (int i = 0; i < D1; ++i) s += xz[i] * CW[(i*D2 + j)*KK + k];
      T1L[z*49 + j*KK + k] = s;
    }
    __syncthreads();

    // ---- T2[k][v][z] = sum_j x2[z,v,j] * T1[z,j,k] ----
    // Thread's z is fixed: cache the whole T1 row + each x row in registers.
    {
      const int z = tid & 15;
      const int v0 = tid >> 4;
      float tr[JK];
#pragma unroll
      for (int jk = 0; jk < JK; ++jk) tr[jk] = T1L[z*49 + jk];
#pragma unroll
      for (int q = 0; q < 4; ++q) {
        const int v = v0 + 16*q;
        const float* xz = &XL[z*S_X + OFF2 + v*D2];
        float xr[D2];
#pragma unroll
        for (int j = 0; j < D2; ++j) xr[j] = xz[j];
#pragma unroll
        for (int k = 0; k < KK; ++k) {
          float s = 0.0f;
#pragma unroll
          for (int j = 0; j < D2; ++j) s += xr[j] * tr[j*KK + k];
          T2L[(k << 10) + (v << 4) + z] = s;
        }
      }
    }

    // ---- pipeline: issue next weight slab, drain current ----
    const int gslab = (p << 6) + u + 1;
    if (gslab < NPATH*64) {
      issue_wl(wts + (size_t)gslab * 4096, WLB + ((u + 1) & 1) * WLSTRIDE, tid);
      wait_async16();   // slab u complete; slab u+1 may remain in flight
    } else {
      wait_async0();    // last slab: nothing to hide behind
    }
    __syncthreads();

    // ---- WMMA: D(16w x 16z) += A(16w x 4uv) * B(4uv x 16z) ----
    if (kq < KK) {   // scalar: this wave has at least one k
      const float* wl = WLB + (u & 1) * WLSTRIDE;
      const int hi = lane >> 4;
      const float* wb0 = &wl[((2*mth)*16 + (lane & 15)) * 65];
      const float* wb1 = wb0 + 16*65;
      const float* bbz = &T2L[lane & 15];

      // A is k-invariant: load once per u-chunk into registers.
      v2f A0r[16], A1r[16];
#pragma unroll
      for (int s4 = 0; s4 < 16; ++s4) {
        const int kb = s4 << 2;
        A0r[s4].x = wb0[kb + 2*hi];  A0r[s4].y = wb0[kb + 2*hi + 1];
        A1r[s4].x = wb1[kb + 2*hi];  A1r[s4].y = wb1[kb + 2*hi + 1];
      }
#pragma unroll
      for (int ai = 0; ai < 2; ++ai) {
        const int k = kq + 4*ai;
        if (ai == 0 || k < KK) {            // ai=0 valid by outer guard
          const float* bb = bbz + (k << 10);
          v8f a0 = acc[0][ai], a1 = acc[1][ai];
#pragma unroll
          for (int s4 = 0; s4 < 16; ++s4) {
            const int kb = s4 << 2;
            v2f B;
            B.x = bb[(kb + hi) << 4];
            B.y = bb[(kb + 2 + hi) << 4];
            a0 = __builtin_amdgcn_wmma_f32_16x16x4_f32(
                     false, A0r[s4], false, B, (short)0, a0, false, false);
            a1 = __builtin_amdgcn_wmma_f32_16x16x4_f32(
                     false, A1r[s4], false, B, (short)0, a1, false, false);
          }
          acc[0][ai] = a0; acc[1][ai] = a1;
        }
      }
    }
    __syncthreads();   // protect T1L/T2L for next u
  }

  // ---- stage result in LDS (padded rows), then coalesced global += ----
  constexpr int FW = 64*KK, SW = FW + 1;
  {
    const int z = lane & 15;
    const int hi = lane >> 4;
#pragma unroll
    for (int ai = 0; ai < 2; ++ai) {
      const int k = kq + 4*ai;
      if (k < KK) {
#pragma unroll
        for (int t = 0; t < 2; ++t) {
          v8f a8 = acc[t][ai];
          const int wt = (2*mth + t) * 16;
#pragma unroll
          for (int r = 0; r < 8; ++r) {
            int m = r + 8*hi;
            T2L[z*SW + (wt + m)*KK + k] = coeff * a8[r];
          }
        }
      }
    }
  }
  __syncthreads();
  for (int e = tid; e < ZT*FW; e += THREADS) {
    int z = e / FW;
    int f = e - z*FW;
    outp[(size_t)(zbase + z)*1024 + OFF3 + f] += T2L[z*SW + f];
  }
  __syncthreads();
}

// ---------------------------------------------------------------------------
// 4 waves per SIMD => 2 blocks per WGP (2 x 129 KB LDS <= 320 KB): caps VGPRs
// at 256/wave (no vgpr-MSB thrash) and preserves cross-block phase overlap.
__launch_bounds__(THREADS, 4)
__global__ void tp_wmma_kernel(const float* __restrict__ xg,
                               const float* __restrict__ wts,
                               const float* __restrict__ w3j,
                               float* __restrict__ outp) {
  __shared__ __align__(16) float XL[ZT * S_X];     // 64.3 KB x tile
  __shared__ float WL[2][WLSTRIDE];                // double-buffered weight slabs
  __shared__ float T2L[16 * 449];                  // B tiles / output staging
  __shared__ float T1L[ZT * 49];
  __shared__ float CW[245];

  const int tid  = threadIdx.x;
  const int lane = tid & 31;
  const int wids = __builtin_amdgcn_readfirstlane(tid >> 5);  // scalar wave id
  const int mth  = wids & 1;           // M-tile pair {2mth, 2mth+1}
  const int kq   = wids >> 1;          // k residue mod 4
  const int zbase = blockIdx.x * ZT;

  // ---- async-copy the 16-sample x tile into LDS ----
  {
    const float* xbase = xg + (size_t)zbase * 1024;
    const unsigned xl0 = (unsigned)(uintptr_t)&XL[0];
#pragma unroll
    for (int it = 0; it < 16; ++it) {
      int slot = it * THREADS + tid;
      int z = slot >> 8;
      int c = slot & 255;
      async_load_b128(xl0 + (unsigned)(z * S_X + c * 4) * 4u,
                      xbase + z * 1024 + c * 4);
    }
  }
  for (int i = tid; i < ZT * 1024; i += THREADS)   // overlap: zero output range
    outp[(size_t)zbase * 1024 + i] = 0.0f;
  wait_async0();
  __syncthreads();

  issue_wl(wts, &WL[0][0], tid);   // prologue: slab (p=0,u=0)

  for (int p = 0; p < NPATH; ++p) {
    switch (p) {
      case  0: run_path<0,0,0>(p, zbase, tid, lane, mth, kq, wts, w3j, outp, XL, &WL[0][0], T2L, T1L, CW); break;
      case  1: run_path<0,1,1>(p, zbase, tid, lane, mth, kq, wts, w3j, outp, XL, &WL[0][0], T2L, T1L, CW); break;
      case  2: run_path<0,2,2>(p, zbase, tid, lane, mth, kq, wts, w3j, outp, XL, &WL[0][0], T2L, T1L, CW); break;
      case  3: run_path<0,3,3>(p, zbase, tid, lane, mth, kq, wts, w3j, outp, XL, &WL[0][0], T2L, T1L, CW); break;
      case  4: run_path<1,0,1>(p, zbase, tid, lane, mth, kq, wts, w3j, outp, XL, &WL[0][0], T2L, T1L, CW); break;
      case  5: run_path<1,1,0>(p, zbase, tid, lane, mth, kq, wts, w3j, outp, XL, &WL[0][0], T2L, T1L, CW); break;
      case  6: run_path<1,1,2>(p, zbase, tid, lane, mth, kq, wts, w3j, outp, XL, &WL[0][0], T2L, T1L, CW); break;
      case  7: run_path<1,2,1>(p, zbase, tid, lane, mth, kq, wts, w3j, outp, XL, &WL[0][0], T2L, T1L, CW); break;
      case  8: run_path<1,2,3>(p, zbase, tid, lane, mth, kq, wts, w3j, outp, XL, &WL[0][0], T2L, T1L, CW); break;
      case  9: run_path<1,3,2>(p, zbase, tid, lane, mth, kq, wts, w3j, outp, XL, &WL[0][0], T2L, T1L, CW); break;
      case 10: run_path<2,0,2>(p, zbase, tid, lane, mth, kq, wts, w3j, outp, XL, &WL[0][0], T2L, T1L, CW); break;
      case 11: run_path<2,1,1>(p, zbase, tid, lane, mth, kq, wts, w3j, outp, XL, &WL[0][0], T2L, T1L, CW); break;
      case 12: run_path<2,1,3>(p, zbase, tid, lane, mth, kq, wts, w3j, outp, XL, &WL[0][0], T2L, T1L, CW); break;
      case 13: run_path<2,2,0>(p, zbase, tid, lane, mth, kq, wts, w3j, outp, XL, &WL[0][0], T2L, T1L, CW); break;
      case 14: run_path<2,2,2>(p, zbase, tid, lane, mth, kq, wts, w3j, outp, XL, &WL[0][0], T2L, T1L, CW); break;
      case 15: run_path<2,3,1>(p, zbase, tid, lane, mth, kq, wts, w3j, outp, XL, &WL[0][0], T2L, T1L, CW); break;
      case 16: run_path<2,3,3>(p, zbase, tid, lane, mth, kq, wts, w3j, outp, XL, &WL[0][0], T2L, T1L, CW); break;
      case 17: run_path<3,0,3>(p, zbase, tid, lane, mth, kq, wts, w3j, outp, XL, &WL[0][0], T2L, T1L, CW); break;
      case 18: run_path<3,1,2>(p, zbase, tid, lane, mth, kq, wts, w3j, outp, XL, &WL[0][0], T2L, T1L, CW); break;
      case 19: run_path<3,2,1>(p, zbase, tid, lane, mth, kq, wts, w3j, outp, XL, &WL[0][0], T2L, T1L, CW); break;
      case 20: run_path<3,2,3>(p, zbase, tid, lane, mth, kq, wts, w3j, outp, XL, &WL[0][0], T2L, T1L, CW); break;
      case 21: run_path<3,3,0>(p, zbase, tid, lane, mth, kq, wts, w3j, outp, XL, &WL[0][0], T2L, T1L, CW); break;
      case 22: run_path<3,3,2>(p, zbase, tid, lane, mth, kq, wts, w3j, outp, XL, &WL[0][0], T2L, T1L, CW); break;
    }
  }
}

// ---------------------------------------------------------------------------
extern "C" void kernel_launch(void* const* d_in, const int* in_sizes, int n_in,
                              void* d_out, int out_size, void* d_ws, size_t ws_size,
                              hipStream_t stream) {
  const float* x   = (const float*)d_in[0];   // (2,1024,1024) f32
  const float* wts = (const float*)d_in[1];   // (23,64,64,64) f32
  float* outp = (float*)d_out;                // (2,1024,1024) f32
  float* w3j  = (float*)d_ws;                 // 1875 floats of Wigner-3j data

  int Z = in_sizes[0] / 1024;                 // 2048
  w3j_init_kernel<<<dim3(1), dim3(32), 0, stream>>>(w3j);
  tp_wmma_kernel<<<dim3(Z / ZT), dim3(THREADS), 0, stream>>>(x, wts, w3j, outp);
}